// ToRGB_5291399709213
// MI455X (gfx1250) — compile-verified
//
#include <hip/hip_runtime.h>
#include <hip/hip_bf16.h>

typedef __attribute__((ext_vector_type(2))) float v2f;
typedef __attribute__((ext_vector_type(8))) float v8f;

#define CH   128
#define OC   3
#define WDIM 512
#define HW   (512 * 512)
#define EPS  1e-8f
#define AFF_SCALE  (1.0f / 22.627416997969522f)   // 1/sqrt(512)
#define CONV_SCALE (1.0f / 11.313708498984761f)   // 1/sqrt(128)

// -------------------------------------------------------------------------
// Kernel 1: style affine + modulate + demodulate, then pack the per-batch
// [3 x 128] weight matrix into WMMA F32_16X16X4 A-operand lane layout:
//   step s (K base = 4s), lane L = h*16 + m (h = L>>4, m = L&15), vgpr v:
//   A-value = w[m][4s + 2h + v]   (zero for m >= 3)
// Stored as aw[b][s][lane][v]  -> 8 * 32 * 32 * 2 floats = 64 KB in d_ws.
// -------------------------------------------------------------------------
__global__ __launch_bounds__(128)
void torgb_modweights(const float* __restrict__ styles,
                      const float* __restrict__ conv_w,
                      const float* __restrict__ aff_w,
                      const float* __restrict__ aff_b,
                      float* __restrict__ aw) {
    const int b = blockIdx.x;       // 0..7
    const int c = threadIdx.x;      // 0..127

    // z[c] = aff_bias[c] + AFF_SCALE * dot(styles[b,:], aff_w[c,:])
    const float* srow = styles + (size_t)b * WDIM;
    const float* arow = aff_w + (size_t)c * WDIM;
    float acc = 0.0f;
#pragma unroll 4
    for (int k = 0; k < WDIM; ++k) acc += srow[k] * arow[k];
    const float z = aff_b[c] + acc * AFF_SCALE;

    float w0 = conv_w[0 * CH + c] * z * CONV_SCALE;
    float w1 = conv_w[1 * CH + c] * z * CONV_SCALE;
    float w2 = conv_w[2 * CH + c] * z * CONV_SCALE;

    // demod[o] = rsqrt(sum_c w[o][c]^2 + eps)
    __shared__ float sq[OC][CH];
    __shared__ float dem[OC];
    sq[0][c] = w0 * w0;
    sq[1][c] = w1 * w1;
    sq[2][c] = w2 * w2;
    __syncthreads();
    if (c < OC) {
        float s = 0.0f;
        for (int i = 0; i < CH; ++i) s += sq[c][i];
        dem[c] = rsqrtf(s + EPS);
    }
    __syncthreads();
    w0 *= dem[0];
    w1 *= dem[1];
    w2 *= dem[2];

    // Pack into A-operand layout.
    const int s_ = c >> 2;        // K-step
    const int r  = c & 3;         // position within the 4-wide K group
    const int h  = r >> 1;        // lane half
    const int v  = r & 1;         // vgpr index
    float* base = aw + ((((size_t)b * 32 + s_) * 32 + h * 16) * 2 + v);
#pragma unroll
    for (int m = 0; m < 16; ++m) {
        float val = (m == 0) ? w0 : (m == 1) ? w1 : (m == 2) ? w2 : 0.0f;
        base[m * 2] = val;
    }
}

// -------------------------------------------------------------------------
// Kernel 2: streaming grouped 1x1 conv via V_WMMA_F32_16X16X4_F32.
// D[o][p] = sum_c W[o][c] * X[c][p]; K = 128 -> 32 WMMA steps per 16-px tile.
// blockIdx.y = batch (scalar base); divergent offsets kept UNSIGNED 32-bit so
// the backend can select SADDR + zext(voffset) global loads instead of
// materializing 64-bit VALU addresses. x streamed once -> NT hints.
// -------------------------------------------------------------------------
__global__ __launch_bounds__(256)
void torgb_conv(const float* __restrict__ x,
                const float* __restrict__ aw,
                const float* __restrict__ bias,
                float* __restrict__ out) {
    const unsigned b     = blockIdx.y;                               // scalar
    const unsigned wslot = blockIdx.x * 8u + (threadIdx.x >> 5);     // 0..511
    const unsigned lane  = threadIdx.x & 31u;
    const unsigned m     = lane & 15u;   // B-matrix pixel column / A row
    const unsigned h     = lane >> 4;    // lane half -> +2 channels

    // Preload all 32 A-operand steps (64 VGPRs), resident for whole kernel.
    v2f a[32];
    const float* awb = aw + (size_t)b * 2048 + (size_t)lane * 2;
#pragma unroll
    for (int s = 0; s < 32; ++s)
        a[s] = *(const v2f*)(awb + s * 64);

    const float* xs = x + (size_t)b * CH * HW;   // uniform (SGPR) base
    float* ob = out + (size_t)b * OC * HW;       // uniform (SGPR) base
    const float b0 = bias[0], b1 = bias[1], b2 = bias[2];

    const unsigned lanebase = 2u * h * HW + m;   // divergent, non-negative
    const unsigned pxbase   = wslot * 512u;

    for (int t = 0; t < 32; ++t) {
        const unsigned off = lanebase + pxbase + (unsigned)t * 16u;
        v8f acc = {0.f, 0.f, 0.f, 0.f, 0.f, 0.f, 0.f, 0.f};
#pragma unroll
        for (int s = 0; s < 32; ++s) {
            const float* xrow = xs + (size_t)(4 * s) * HW;  // uniform per step
            v2f bx;
            bx.x = __builtin_nontemporal_load(xrow + off);            // 4s+2h
            bx.y = __builtin_nontemporal_load(xrow + off + (unsigned)HW);
            acc = __builtin_amdgcn_wmma_f32_16x16x4_f32(
                false, a[s], false, bx, (short)0, acc, false, false);
        }
        // D rows 0..2 live in acc[0..2] on lanes 0-15 (N = m).
        if (h == 0u) {
            const unsigned p = pxbase + (unsigned)t * 16u + m;
            __builtin_nontemporal_store(acc[0] + b0, ob + p);
            __builtin_nontemporal_store(acc[1] + b1, ob + p + (unsigned)HW);
            __builtin_nontemporal_store(acc[2] + b2, ob + p + 2u * (unsigned)HW);
        }
    }
}

extern "C" void kernel_launch(void* const* d_in, const int* in_sizes, int n_in,
                              void* d_out, int out_size, void* d_ws, size_t ws_size,
                              hipStream_t stream) {
    const float* x      = (const float*)d_in[0];  // [8,128,512,512]
    const float* styles = (const float*)d_in[1];  // [8,512]
    const float* conv_w = (const float*)d_in[2];  // [3,128]
    const float* conv_b = (const float*)d_in[3];  // [3]
    const float* aff_w  = (const float*)d_in[4];  // [128,512]
    const float* aff_b  = (const float*)d_in[5];  // [128]
    float* out = (float*)d_out;                   // [8,3,512,512]
    float* aw  = (float*)d_ws;                    // 64 KB packed A-operands

    torgb_modweights<<<8, 128, 0, stream>>>(styles, conv_w, aff_w, aff_b, aw);
    dim3 grid(64, 8);
    torgb_conv<<<grid, 256, 0, stream>>>(x, aw, conv_b, out);
}